// MARN_20504173871339
// MI455X (gfx1250) — compile-verified
//
#include <hip/hip_runtime.h>
#include <hip/hip_bf16.h>
#include <math.h>

// ---------------------------------------------------------------------------
// MARN on MI455X (gfx1250): fp32 WMMA (V_WMMA_F32_16X16X4_F32) everywhere.
// Phase 1: 3 transformer encoders (LDS-staged WMMA GEMMs + fused attention).
// Phase 2: 256-step recurrent scan fused into ONE persistent 32-wave kernel,
//          with all weights pre-transposed so B-fragments are contiguous pairs.
// All fragment pairs are 8-byte aligned (even strides) so every pair load is a
// single ds_load_b64 / global_load_b64 with immediate offsets.
// ---------------------------------------------------------------------------

#define T_STEPS 256
#define N_BATCH 128
#define D_TOT   712
#define DM_L    100
#define DM_A    100
#define DM_V    512
#define NHEAD   8
#define DKV     40
#define HID     320      // NHEAD*DKV
#define D_INNER 50
#define DH_L    128
#define DH_A    16
#define DH_V    64
#define TOT_H   208
#define RL_N    16
#define RA_N    128
#define RV_N    100
#define RSUM    244
#define MAP_H   64
#define H_OUT   32
#define OUT_DIM 6
#define MROWS   (N_BATCH * T_STEPS)   // 32768

// LDS strides (even -> 8B-aligned pairs; proven bank-conflict-free for frags)
#define AS_STR  34
#define KS_STR  42
#define SW_STR  258

typedef __attribute__((ext_vector_type(2))) float v2f;
typedef __attribute__((ext_vector_type(8))) float v8f;

__device__ __forceinline__ int imin(int a, int b) { return a < b ? a : b; }

__device__ __forceinline__ v8f wmma4(v2f a, v2f b, v8f c) {
  return __builtin_amdgcn_wmma_f32_16x16x4_f32(false, a, false, b, (short)0, c,
                                               false, false);
}
__device__ __forceinline__ v8f v8zero() {
  v8f z;
#pragma unroll
  for (int i = 0; i < 8; ++i) z[i] = 0.f;
  return z;
}
// p is 8-byte aligned at every call site (even strides + even k)
__device__ __forceinline__ v2f ldpair(const float* p) {
  return *(const v2f*)p;
}
__device__ __forceinline__ float warp_sum(float v) {
#pragma unroll
  for (int o = 16; o; o >>= 1) v += __shfl_xor(v, o, 32);
  return v;
}
__device__ __forceinline__ float warp_max(float v) {
#pragma unroll
  for (int o = 16; o; o >>= 1) v = fmaxf(v, __shfl_xor(v, o, 32));
  return v;
}
__device__ __forceinline__ float sigm(float x) { return 1.f / (1.f + __expf(-x)); }

// --------------------------- slice / transpose -----------------------------
__global__ void slice_kernel(const float* __restrict__ x, float* __restrict__ xm,
                             int dm, int off) {
  int idx = blockIdx.x * blockDim.x + threadIdx.x;
  if (idx >= MROWS * dm) return;
  int k = idx % dm;
  int m = idx / dm;
  int t = m % T_STEPS;
  int n = m / T_STEPS;
  xm[idx] = x[(t * N_BATCH + n) * D_TOT + off + k];
}

// WT[col*K + k] = W[k*Nout + col]  (one-time, tiny)
__global__ void transpose_kernel(const float* __restrict__ W,
                                 float* __restrict__ WT, int K, int Nout) {
  int idx = blockIdx.x * blockDim.x + threadIdx.x;
  if (idx >= K * Nout) return;
  int k = idx / Nout, c = idx % Nout;
  WT[(size_t)c * K + k] = W[idx];
}

// --------------------------- generic WMMA GEMM -----------------------------
// Y[M,Nout] = act( A[M,K] @ W[K,Nout] + bias + res )
// Block 256 thr (8 waves). Tile 128 rows x 64 cols, BK=32; A and B staged in
// LDS (B k-transposed, even strides); inner loop = 5 ds_load_b64 + 4 WMMAs,
// software-pipelined one k-step ahead.
__global__ void __launch_bounds__(256) gemm_kernel(
    const float* __restrict__ Am, const float* __restrict__ Wm,
    const float* __restrict__ bias, const float* __restrict__ res,
    float* __restrict__ Y, int K, int Nout, int act) {
  __shared__ float As[128 * AS_STR];  // As[row*34 + k]
  __shared__ float Bs[64 * AS_STR];   // Bs[col*34 + k]
  const int m0 = blockIdx.x * 128;
  const int n0 = blockIdx.y * 64;
  const int wave = threadIdx.x >> 5, lane = threadIdx.x & 31;
  const int l15 = lane & 15, kh = (lane >> 4) * 2;
  const int rt = wave;
  const int Kc = K - 1;

  v8f acc[4];
#pragma unroll
  for (int c = 0; c < 4; ++c) acc[c] = v8zero();

  for (int k0 = 0; k0 < K; k0 += 32) {
    // stage A chunk (k-overrun zeroed: it feeds the reduction)
    for (int i = threadIdx.x; i < 128 * 32; i += 256) {
      int r = i >> 5, c = i & 31;
      int gk = k0 + c;
      float msk = (gk <= Kc) ? 1.f : 0.f;
      As[r * AS_STR + c] = Am[(size_t)(m0 + r) * K + imin(gk, Kc)] * msk;
    }
    // stage B chunk transposed (col-overrun columns never stored -> clamp only)
    for (int i = threadIdx.x; i < 32 * 64; i += 256) {
      int kk = i >> 6, col = i & 63;  // col fastest -> coalesced global reads
      int gk = k0 + kk;
      float msk = (gk <= Kc) ? 1.f : 0.f;
      int gc = imin(n0 + col, Nout - 1);
      Bs[col * AS_STR + kk] = Wm[(size_t)imin(gk, Kc) * Nout + gc] * msk;
    }
    __syncthreads();
    const float* arow = &As[(rt * 16 + l15) * AS_STR];
    const float* bcol = &Bs[l15 * AS_STR];
    int rem = K - k0;
    if (rem >= 32) {
      // depth-1 software pipeline, fully unrolled (guards fold at compile time)
      v2f a = ldpair(arow + kh);
      v2f b0 = ldpair(bcol + 0 * 16 * AS_STR + kh);
      v2f b1 = ldpair(bcol + 1 * 16 * AS_STR + kh);
      v2f b2 = ldpair(bcol + 2 * 16 * AS_STR + kh);
      v2f b3 = ldpair(bcol + 3 * 16 * AS_STR + kh);
#pragma unroll
      for (int ks = 0; ks < 8; ++ks) {
        v2f an, bn0, bn1, bn2, bn3;
        if (ks < 7) {
          int kn = (ks + 1) * 4 + kh;
          an = ldpair(arow + kn);
          bn0 = ldpair(bcol + 0 * 16 * AS_STR + kn);
          bn1 = ldpair(bcol + 1 * 16 * AS_STR + kn);
          bn2 = ldpair(bcol + 2 * 16 * AS_STR + kn);
          bn3 = ldpair(bcol + 3 * 16 * AS_STR + kn);
        }
        acc[0] = wmma4(a, b0, acc[0]);
        acc[1] = wmma4(a, b1, acc[1]);
        acc[2] = wmma4(a, b2, acc[2]);
        acc[3] = wmma4(a, b3, acc[3]);
        if (ks < 7) { a = an; b0 = bn0; b1 = bn1; b2 = bn2; b3 = bn3; }
      }
      // pin schedule: preload DS x5, then 7x (DS x5, WMMA x4), final WMMA x4
      __builtin_amdgcn_sched_group_barrier(0x100, 5, 0);
#pragma unroll
      for (int i = 0; i < 7; ++i) {
        __builtin_amdgcn_sched_group_barrier(0x100, 5, 0);
        __builtin_amdgcn_sched_group_barrier(0x008, 4, 0);
      }
      __builtin_amdgcn_sched_group_barrier(0x008, 4, 0);
    } else {
      int ksteps = (rem + 3) >> 2;
      for (int ks = 0; ks < ksteps; ++ks) {
        int kk = ks * 4 + kh;
        v2f a = ldpair(arow + kk);
#pragma unroll
        for (int ct = 0; ct < 4; ++ct) {
          v2f b = ldpair(bcol + ct * 16 * AS_STR + kk);
          acc[ct] = wmma4(a, b, acc[ct]);
        }
      }
    }
    __syncthreads();
  }
#pragma unroll
  for (int ct = 0; ct < 4; ++ct) {
    int col = n0 + ct * 16 + l15;
    if (col >= Nout) continue;
    float bv = bias ? bias[col] : 0.f;
#pragma unroll
    for (int j = 0; j < 8; ++j) {
      int row = m0 + rt * 16 + j + ((lane >> 4) << 3);
      size_t idx = (size_t)row * Nout + col;
      float v = acc[ct][j] + bv;
      if (res) v += res[idx];
      if (act) v = fmaxf(v, 0.f);
      Y[idx] = v;
    }
  }
}

// --------------------------- layer norm (in-place) -------------------------
__global__ void __launch_bounds__(256) ln_kernel(float* __restrict__ Y,
                                                 const float* __restrict__ g,
                                                 const float* __restrict__ b,
                                                 int dm) {
  int row = blockIdx.x * 8 + (threadIdx.x >> 5);
  int lane = threadIdx.x & 31;
  float s1 = 0.f, s2 = 0.f;
  for (int c = lane; c < dm; c += 32) {
    float v = Y[(size_t)row * dm + c];
    s1 += v; s2 += v * v;
  }
  s1 = warp_sum(s1); s2 = warp_sum(s2);
  float mean = s1 / dm;
  float var = s2 / dm - mean * mean;
  float inv = rsqrtf(var + 1e-6f);
  for (int c = lane; c < dm; c += 32) {
    size_t i = (size_t)row * dm + c;
    Y[i] = (Y[i] - mean) * inv * g[c] + b[c];
  }
}

// --------------------------- fused attention -------------------------------
// One block per (n, head): S = softmax(Q K^T / sqrt(40)), O = S V. T=256, d=40.
__global__ void __launch_bounds__(256) attn_kernel(const float* __restrict__ Q,
                                                   const float* __restrict__ Kp,
                                                   const float* __restrict__ Vp,
                                                   float* __restrict__ O) {
  extern __shared__ float sm[];
  float* Ks = sm;                        // 256*42 (row-major, even pad)
  float* Vt = Ks + 256 * KS_STR;         // 40*258 (transposed: Vt[col*258+key])
  float* Qs = Vt + DKV * SW_STR;         // 8 * 16*42
  float* Ss = Qs + 8 * 16 * KS_STR;      // 8 * 16*258
  const int nh = blockIdx.x;
  const int n = nh >> 3, h = nh & 7;
  const size_t base = (size_t)n * T_STEPS * HID + h * DKV;

  for (int i = threadIdx.x; i < 256 * DKV; i += 256) {
    int key = i / DKV, c = i % DKV;
    Ks[key * KS_STR + c] = Kp[base + (size_t)key * HID + c];
    Vt[c * SW_STR + key] = Vp[base + (size_t)key * HID + c];
  }
  __syncthreads();

  const int wave = threadIdx.x >> 5, lane = threadIdx.x & 31;
  const int l15 = lane & 15, kh = (lane >> 4) * 2;
  float* Sw = Ss + wave * 16 * SW_STR;
  float* Qw = Qs + wave * 16 * KS_STR;
  const float scale = rsqrtf((float)DKV);

  for (int qt = wave; qt < 16; qt += 8) {
    const int q0 = qt * 16;
    for (int i = lane; i < 16 * DKV; i += 32) {
      int r = i / DKV, c = i % DKV;
      Qw[r * KS_STR + c] = Q[base + (size_t)(q0 + r) * HID + c];
    }
    // scores S[16,256]
    const float* qrow = Qw + l15 * KS_STR;
    for (int kt = 0; kt < 16; ++kt) {
      const float* krow = Ks + (kt * 16 + l15) * KS_STR;
      v8f acc = v8zero();
      v2f a = ldpair(qrow + kh);
      v2f b = ldpair(krow + kh);
#pragma unroll
      for (int ks = 0; ks < 10; ++ks) {  // K = 40, pipelined
        v2f an, bn;
        if (ks < 9) {
          int kn = (ks + 1) * 4 + kh;
          an = ldpair(qrow + kn);
          bn = ldpair(krow + kn);
        }
        acc = wmma4(a, b, acc);
        if (ks < 9) { a = an; b = bn; }
      }
#pragma unroll
      for (int j = 0; j < 8; ++j) {
        int r = j + ((lane >> 4) << 3);
        Sw[r * SW_STR + kt * 16 + l15] = acc[j] * scale;
      }
    }
    // softmax over 256 keys, per row
    for (int r = 0; r < 16; ++r) {
      float mx = -1e30f;
      for (int c = lane; c < 256; c += 32) mx = fmaxf(mx, Sw[r * SW_STR + c]);
      mx = warp_max(mx);
      float s = 0.f;
      for (int c = lane; c < 256; c += 32) {
        float e = __expf(Sw[r * SW_STR + c] - mx);
        Sw[r * SW_STR + c] = e; s += e;
      }
      s = warp_sum(s);
      float inv = 1.f / s;
      for (int c = lane; c < 256; c += 32) Sw[r * SW_STR + c] *= inv;
    }
    // O = P @ V  (K = 256, Nout = 40); col clamp only (guarded store)
    const float* srow = Sw + l15 * SW_STR;
    for (int ct = 0; ct < 3; ++ct) {
      int col = ct * 16 + l15;
      const float* vcol = Vt + imin(col, DKV - 1) * SW_STR;
      v8f acc = v8zero();
      v2f a = ldpair(srow + kh);
      v2f b = ldpair(vcol + kh);
#pragma unroll 4
      for (int ks = 0; ks < 63; ++ks) {  // peel last iteration
        int kn = (ks + 1) * 4 + kh;
        v2f an = ldpair(srow + kn);
        v2f bn = ldpair(vcol + kn);
        acc = wmma4(a, b, acc);
        a = an; b = bn;
      }
      acc = wmma4(a, b, acc);
      if (col < DKV) {
#pragma unroll
        for (int j = 0; j < 8; ++j) {
          int r = j + ((lane >> 4) << 3);
          O[base + (size_t)(q0 + r) * HID + col] = acc[j];
        }
      }
    }
  }
}

// --------------------------- persistent scan kernel ------------------------
struct ScanArgs {
  const float *xl, *xa, *xv;  // encoder outputs, [n*T+t, dm]
  const float *WlT, *UlT, *VlT, *Wl_b, *Ul_b, *Vl_b;
  const float *WaT, *UaT, *VaT, *Wa_b, *Ua_b, *Va_b;
  const float *WvT, *UvT, *VvT, *Wv_b, *Uv_b, *Vv_b;
  const float *attT, *att_b;
  const float *rlT, *rl_b, *raT, *ra_b, *rvT, *rv_b;
  const float *fc1T, *fc1_b, *fc2T, *fc2_b;
  const float *o1T, *o1_b, *o2_w, *o2_b;
  float *cl[2], *hl[2], *ca[2], *ha[2], *cv[2], *hv[2], *z[2];
  float *attbuf, *red, *t1, *u;
  float *out;
};

__global__ void __launch_bounds__(1024, 1) scan_kernel(ScanArgs P) {
  const int tid = threadIdx.x;
  const int wave = tid >> 5, lane = tid & 31;
  const int l15 = lane & 15, kh = (lane >> 4) * 2;

  // zero-init states (buffer 0)
  for (int i = tid; i < 128 * 128; i += 1024) { P.cl[0][i] = 0.f; P.hl[0][i] = 0.f; }
  for (int i = tid; i < 128 * 16; i += 1024)  { P.ca[0][i] = 0.f; P.ha[0][i] = 0.f; }
  for (int i = tid; i < 128 * 64; i += 1024)  { P.cv[0][i] = 0.f; P.hv[0][i] = 0.f; }
  for (int i = tid; i < 128 * TOT_H; i += 1024) P.z[0][i] = 0.f;
  __syncthreads();

  for (int t = 0; t < T_STEPS; ++t) {
    const int cur = t & 1, nxt = cur ^ 1;

    // ---- S1: three LSTHM cells, gates fused (f,i,o,ch accumulators) ------
    for (int task = wave; task < 104; task += 32) {
      int C, din, rt, ct;
      const float *xin, *WT, *UT, *VT, *Wb, *Ub, *Vb, *hOld, *cOld;
      float *cNew, *hNew;
      if (task < 64) {
        C = DH_L; din = DM_L; rt = task >> 3; ct = task & 7;
        xin = P.xl; WT = P.WlT; UT = P.UlT; VT = P.VlT;
        Wb = P.Wl_b; Ub = P.Ul_b; Vb = P.Vl_b;
        hOld = P.hl[cur]; cOld = P.cl[cur]; cNew = P.cl[nxt]; hNew = P.hl[nxt];
      } else if (task < 72) {
        int q = task - 64;
        C = DH_A; din = DM_A; rt = q; ct = 0;
        xin = P.xa; WT = P.WaT; UT = P.UaT; VT = P.VaT;
        Wb = P.Wa_b; Ub = P.Ua_b; Vb = P.Va_b;
        hOld = P.ha[cur]; cOld = P.ca[cur]; cNew = P.ca[nxt]; hNew = P.ha[nxt];
      } else {
        int q = task - 72;
        C = DH_V; din = DM_V; rt = q >> 2; ct = q & 3;
        xin = P.xv; WT = P.WvT; UT = P.UvT; VT = P.VvT;
        Wb = P.Wv_b; Ub = P.Uv_b; Vb = P.Vv_b;
        hOld = P.hv[cur]; cOld = P.cv[cur]; cNew = P.cv[nxt]; hNew = P.hv[nxt];
      }
      const int Ktot = din + C + TOT_H;  // divisible by 4; boundaries even
      const int nA = rt * 16 + l15;
      const float* zc = P.z[cur];
      const float* xrow = xin + (size_t)(nA * T_STEPS + t) * din;
      int col0 = ct * 16 + l15;
      const float* wb0[4]; const float* ub0[4]; const float* vb0[4];
#pragma unroll
      for (int g = 0; g < 4; ++g) {
        int col = g * C + col0;
        wb0[g] = WT + (size_t)col * din;
        ub0[g] = UT + (size_t)col * C;
        vb0[g] = VT + (size_t)col * TOT_H;
      }
      auto selA = [&](int k) -> const float* {
        return (k < din) ? (xrow + k)
             : (k < din + C) ? (hOld + nA * C + (k - din))
                             : (zc + nA * TOT_H + (k - din - C));
      };
      v8f acc[4];
#pragma unroll
      for (int g = 0; g < 4; ++g) acc[g] = v8zero();
      const int nks = Ktot >> 2;
      // depth-1 pipeline, last iteration peeled
      v2f a = ldpair(selA(kh));
      v2f b0, b1, b2, b3;
      {
        int k = kh;
        b0 = ldpair((k < din) ? wb0[0] + k : (k < din + C) ? ub0[0] + k - din : vb0[0] + k - din - C);
        b1 = ldpair((k < din) ? wb0[1] + k : (k < din + C) ? ub0[1] + k - din : vb0[1] + k - din - C);
        b2 = ldpair((k < din) ? wb0[2] + k : (k < din + C) ? ub0[2] + k - din : vb0[2] + k - din - C);
        b3 = ldpair((k < din) ? wb0[3] + k : (k < din + C) ? ub0[3] + k - din : vb0[3] + k - din - C);
      }
      for (int ks = 0; ks < nks - 1; ++ks) {
        int kn = (ks + 1) * 4 + kh;
        v2f an = ldpair(selA(kn));
        v2f bn0 = ldpair((kn < din) ? wb0[0] + kn : (kn < din + C) ? ub0[0] + kn - din : vb0[0] + kn - din - C);
        v2f bn1 = ldpair((kn < din) ? wb0[1] + kn : (kn < din + C) ? ub0[1] + kn - din : vb0[1] + kn - din - C);
        v2f bn2 = ldpair((kn < din) ? wb0[2] + kn : (kn < din + C) ? ub0[2] + kn - din : vb0[2] + kn - din - C);
        v2f bn3 = ldpair((kn < din) ? wb0[3] + kn : (kn < din + C) ? ub0[3] + kn - din : vb0[3] + kn - din - C);
        acc[0] = wmma4(a, b0, acc[0]);
        acc[1] = wmma4(a, b1, acc[1]);
        acc[2] = wmma4(a, b2, acc[2]);
        acc[3] = wmma4(a, b3, acc[3]);
        a = an; b0 = bn0; b1 = bn1; b2 = bn2; b3 = bn3;
      }
      acc[0] = wmma4(a, b0, acc[0]);
      acc[1] = wmma4(a, b1, acc[1]);
      acc[2] = wmma4(a, b2, acc[2]);
      acc[3] = wmma4(a, b3, acc[3]);

      float bg[4];
#pragma unroll
      for (int g = 0; g < 4; ++g)
        bg[g] = Wb[g * C + col0] + Ub[g * C + col0] + Vb[g * C + col0];
#pragma unroll
      for (int j = 0; j < 8; ++j) {
        int n = rt * 16 + j + ((lane >> 4) << 3);
        float f = sigm(acc[0][j] + bg[0]);
        float ig = sigm(acc[1][j] + bg[1]);
        float og = sigm(acc[2][j] + bg[2]);
        float ch = tanhf(acc[3][j] + bg[3]);
        float cv_ = f * cOld[n * C + col0] + ig * ch;
        cNew[n * C + col0] = cv_;
        hNew[n * C + col0] = tanhf(cv_) * og;
      }
    }
    __syncthreads();

    const float* clN = P.cl[nxt];
    const float* caN = P.ca[nxt];
    const float* cvN = P.cv[nxt];

    // ---- S2: logits = new_cs @ att_w + att_b  [128, 832] ------------------
    for (int task = wave; task < 8 * 52; task += 32) {
      int rt = task / 52, ct = task % 52;
      int nA = rt * 16 + l15, col = ct * 16 + l15;
      const float* pb0 = P.attT + (size_t)col * TOT_H;
      auto selA = [&](int k) -> const float* {
        return (k < 128) ? (clN + nA * 128 + k)
             : (k < 144) ? (caN + nA * 16 + k - 128)
                         : (cvN + nA * 64 + k - 144);
      };
      v8f acc = v8zero();
      v2f a = ldpair(selA(kh));
      v2f b = ldpair(pb0 + kh);
      for (int ks = 0; ks < 51; ++ks) {  // K = 208, peel last
        int kn = (ks + 1) * 4 + kh;
        v2f an = ldpair(selA(kn));
        v2f bn = ldpair(pb0 + kn);
        acc = wmma4(a, b, acc);
        a = an; b = bn;
      }
      acc = wmma4(a, b, acc);
      float bb = P.att_b[col];
#pragma unroll
      for (int j = 0; j < 8; ++j) {
        int n = rt * 16 + j + ((lane >> 4) << 3);
        P.attbuf[n * 832 + col] = acc[j] + bb;
      }
    }
    __syncthreads();

    // ---- S3: softmax per (n, att-group) then multiply by cs ---------------
    for (int grp = wave; grp < 512; grp += 32) {
      int n = grp >> 2, g = grp & 3;
      float* p = P.attbuf + n * 832 + g * TOT_H;
      float mx = -1e30f;
      for (int c = lane; c < TOT_H; c += 32) mx = fmaxf(mx, p[c]);
      mx = warp_max(mx);
      float s = 0.f;
      for (int c = lane; c < TOT_H; c += 32) {
        float e = __expf(p[c] - mx);
        p[c] = e; s += e;
      }
      s = warp_sum(s);
      float inv = 1.f / s;
      for (int c = lane; c < TOT_H; c += 32) {
        float csv = (c < 128) ? clN[n * 128 + c]
                    : (c < 144) ? caN[n * 16 + c - 128] : cvN[n * 64 + c - 144];
        p[c] = p[c] * inv * csv;
      }
    }
    __syncthreads();

    // ---- S4: reductions red = [rl(16) | ra(128) | rv(100)] [128,244] ------
    for (int task = wave; task < 128; task += 32) {
      int mode, rt, ct;
      if (task < 8) { mode = 0; rt = task; ct = 0; }
      else if (task < 72) { int q = task - 8; mode = 1; rt = q >> 3; ct = q & 7; }
      else { int q = task - 72; mode = 2; rt = q / 7; ct = q % 7; }
      const int Kd   = (mode == 0) ? 512 : (mode == 1) ? 64 : 256;
      const int Nout = (mode == 0) ? RL_N : (mode == 1) ? RA_N : RV_N;
      const int cbase = (mode == 0) ? 0 : (mode == 1) ? 16 : 144;
      const int jsh  = (mode == 0) ? 7 : (mode == 1) ? 4 : 6;
      const int jmask = (1 << jsh) - 1;
      const int joff = (mode == 0) ? 0 : (mode == 1) ? 128 : 144;
      const float* WpT = (mode == 0) ? P.rlT : (mode == 1) ? P.raT : P.rvT;
      const float* bp = (mode == 0) ? P.rl_b : (mode == 1) ? P.ra_b : P.rv_b;
      int nA = rt * 16 + l15, col = ct * 16 + l15;
      const float* pb0 = WpT + (size_t)imin(col, Nout - 1) * Kd;
      const float* pa0 = P.attbuf + nA * 832 + joff;
      v8f acc = v8zero();
      v2f a = ldpair(pa0 + (kh >> jsh) * TOT_H + (kh & jmask));
      v2f b = ldpair(pb0 + kh);
      const int nks = Kd >> 2;
      for (int ks = 0; ks < nks - 1; ++ks) {
        int kn = (ks + 1) * 4 + kh;
        v2f an = ldpair(pa0 + (kn >> jsh) * TOT_H + (kn & jmask));
        v2f bn = ldpair(pb0 + kn);
        acc = wmma4(a, b, acc);
        a = an; b = bn;
      }
      acc = wmma4(a, b, acc);
      if (col < Nout) {
        float bb = bp[col];
#pragma unroll
        for (int j = 0; j < 8; ++j) {
          int n = rt * 16 + j + ((lane >> 4) << 3);
          P.red[n * RSUM + cbase + col] = acc[j] + bb;
        }
      }
    }
    __syncthreads();

    // ---- S5: t1 = relu(red @ fc1 + b)  [128,64] ---------------------------
    {
      int task = wave;  // exactly 32 tasks: 8 rt x 4 ct
      int rt = task >> 2, ct = task & 3;
      int nA = rt * 16 + l15, col = ct * 16 + l15;
      const float* pb0 = P.fc1T + (size_t)col * RSUM;
      const float* pa0 = P.red + nA * RSUM;
      v8f acc = v8zero();
      v2f a = ldpair(pa0 + kh);
      v2f b = ldpair(pb0 + kh);
      for (int ks = 0; ks < 60; ++ks) {  // K = 244, peel last
        int kn = (ks + 1) * 4 + kh;
        v2f an = ldpair(pa0 + kn);
        v2f bn = ldpair(pb0 + kn);
        acc = wmma4(a, b, acc);
        a = an; b = bn;
      }
      acc = wmma4(a, b, acc);
      float bb = P.fc1_b[col];
#pragma unroll
      for (int j = 0; j < 8; ++j) {
        int n = rt * 16 + j + ((lane >> 4) << 3);
        P.t1[n * MAP_H + col] = fmaxf(acc[j] + bb, 0.f);
      }
    }
    __syncthreads();

    // ---- S6: z_next = t1 @ fc2 + b  [128,208] -----------------------------
    float* zN = P.z[nxt];
    for (int task = wave; task < 104; task += 32) {  // 8 rt x 13 ct
      int rt = task / 13, ct = task % 13;
      int nA = rt * 16 + l15, col = ct * 16 + l15;
      const float* pb0 = P.fc2T + (size_t)col * MAP_H;
      const float* pa0 = P.t1 + nA * MAP_H;
      v8f acc = v8zero();
      v2f a = ldpair(pa0 + kh);
      v2f b = ldpair(pb0 + kh);
#pragma unroll
      for (int ks = 0; ks < 15; ++ks) {  // K = 64, peel last
        int kn = (ks + 1) * 4 + kh;
        v2f an = ldpair(pa0 + kn);
        v2f bn = ldpair(pb0 + kn);
        acc = wmma4(a, b, acc);
        a = an; b = bn;
      }
      acc = wmma4(a, b, acc);
      float bb = P.fc2_b[col];
#pragma unroll
      for (int j = 0; j < 8; ++j) {
        int n = rt * 16 + j + ((lane >> 4) << 3);
        zN[n * TOT_H + col] = acc[j] + bb;
      }
    }
    __syncthreads();

    // ---- S7: u = relu([h_l,h_a,h_v,z] @ o1 + b)  [128,32] -----------------
    const float* hlN = P.hl[nxt];
    const float* haN = P.ha[nxt];
    const float* hvN = P.hv[nxt];
    for (int task = wave; task < 16; task += 32) {  // 8 rt x 2 ct
      int rt = task >> 1, ct = task & 1;
      int nA = rt * 16 + l15, col = ct * 16 + l15;
      const float* pb0 = P.o1T + (size_t)col * 416;
      auto selA = [&](int k) -> const float* {
        return (k < 128) ? (hlN + nA * 128 + k)
             : (k < 144) ? (haN + nA * 16 + k - 128)
             : (k < 208) ? (hvN + nA * 64 + k - 144)
                         : (zN + nA * TOT_H + k - 208);
      };
      v8f acc = v8zero();
      v2f a = ldpair(selA(kh));
      v2f b = ldpair(pb0 + kh);
      for (int ks = 0; ks < 103; ++ks) {  // K = 416, peel last
        int kn = (ks + 1) * 4 + kh;
        v2f an = ldpair(selA(kn));
        v2f bn = ldpair(pb0 + kn);
        acc = wmma4(a, b, acc);
        a = an; b = bn;
      }
      acc = wmma4(a, b, acc);
      float bb = P.o1_b[col];
#pragma unroll
      for (int j = 0; j < 8; ++j) {
        int n = rt * 16 + j + ((lane >> 4) << 3);
        P.u[n * H_OUT + col] = fmaxf(acc[j] + bb, 0.f);
      }
    }
    __syncthreads();

    // ---- S8: out = softmax(u @ o2 + b)  [128,6] ---------------------------
    if (tid < N_BATCH) {
      int n = tid;
      float lg[OUT_DIM];
#pragma unroll
      for (int j = 0; j < OUT_DIM; ++j) {
        float s = P.o2_b[j];
        for (int k = 0; k < H_OUT; ++k)
          s += P.u[n * H_OUT + k] * P.o2_w[k * OUT_DIM + j];
        lg[j] = s;
      }
      float mx = lg[0];
#pragma unroll
      for (int j = 1; j < OUT_DIM; ++j) mx = fmaxf(mx, lg[j]);
      float ssum = 0.f;
#pragma unroll
      for (int j = 0; j < OUT_DIM; ++j) { lg[j] = __expf(lg[j] - mx); ssum += lg[j]; }
      float inv = 1.f / ssum;
#pragma unroll
      for (int j = 0; j < OUT_DIM; ++j)
        P.out[(size_t)(t * N_BATCH + n) * OUT_DIM + j] = lg[j] * inv;
    }
    __syncthreads();
  }
}

// ---------------------------------------------------------------------------
// Host side
// ---------------------------------------------------------------------------
struct EncP {
  const float *wq_w, *wq_b, *wk_w, *wk_b, *wv_w, *wv_b, *fc_w, *fc_b;
  const float *w1_w, *w1_b, *w2_w, *w2_b, *ln1_g, *ln1_b, *ln2_g, *ln2_b;
};
struct CellP { const float *Ww, *Wb, *Uw, *Ub, *Vw, *Vb; };

extern "C" void kernel_launch(void* const* d_in, const int* in_sizes, int n_in,
                              void* d_out, int out_size, void* d_ws, size_t ws_size,
                              hipStream_t stream) {
  (void)out_size; (void)ws_size;
  // ----- locate x and detect flattening scheme -----
  const int X_SIZE = T_STEPS * N_BATCH * D_TOT;  // 23,330,816
  int xi = -1;
  for (int i = 0; i < n_in; ++i)
    if (in_sizes[i] == X_SIZE) { xi = i; break; }
  int first = (xi == 0) ? 1 : 0;
  bool sorted_scheme = (xi >= 0 && in_sizes[first] == 832);  // att.b first

  int cursor = 0;
  auto nextp = [&]() -> const float* {
    if (cursor == xi) ++cursor;
    return (const float*)d_in[cursor++];
  };

  const float* xptr = (xi >= 0) ? (const float*)d_in[xi] : nullptr;
  EncP el, ea, ev;
  CellP cl_, ca_, cv_;
  const float *att_w, *att_b, *rl_w, *rl_b, *ra_w, *ra_b, *rv_w, *rv_b;
  const float *fc1_w, *fc1_b, *fc2_w, *fc2_b, *o1_w, *o1_b, *o2_w, *o2_b;

  if (!sorted_scheme) {
    auto getEnc = [&](EncP& e) {
      e.wq_w = nextp(); e.wq_b = nextp(); e.wk_w = nextp(); e.wk_b = nextp();
      e.wv_w = nextp(); e.wv_b = nextp(); e.fc_w = nextp(); e.fc_b = nextp();
      e.w1_w = nextp(); e.w1_b = nextp(); e.w2_w = nextp(); e.w2_b = nextp();
      e.ln1_g = nextp(); e.ln1_b = nextp(); e.ln2_g = nextp(); e.ln2_b = nextp();
    };
    auto getCell = [&](CellP& c) {
      c.Ww = nextp(); c.Wb = nextp(); c.Uw = nextp(); c.Ub = nextp();
      c.Vw = nextp(); c.Vb = nextp();
    };
    getEnc(el); getEnc(ea); getEnc(ev);
    getCell(cl_); getCell(ca_); getCell(cv_);
    att_w = nextp(); att_b = nextp();
    rl_w = nextp(); rl_b = nextp(); ra_w = nextp(); ra_b = nextp();
    rv_w = nextp(); rv_b = nextp();
    fc1_w = nextp(); fc1_b = nextp(); fc2_w = nextp(); fc2_b = nextp();
    o1_w = nextp(); o1_b = nextp(); o2_w = nextp(); o2_b = nextp();
  } else {
    att_b = nextp(); att_w = nextp();
    auto getEncS = [&](EncP& e) {
      e.fc_b = nextp(); e.fc_w = nextp();
      e.ln1_b = nextp(); e.ln1_g = nextp();
      e.ln2_b = nextp(); e.ln2_g = nextp();
      e.w1_b = nextp(); e.w1_w = nextp();
      e.w2_b = nextp(); e.w2_w = nextp();
      e.wk_b = nextp(); e.wk_w = nextp();
      e.wq_b = nextp(); e.wq_w = nextp();
      e.wv_b = nextp(); e.wv_w = nextp();
    };
    getEncS(ea); getEncS(el); getEncS(ev);
    fc1_b = nextp(); fc1_w = nextp(); fc2_b = nextp(); fc2_w = nextp();
    auto getCellS = [&](CellP& c) {
      c.Ub = nextp(); c.Uw = nextp(); c.Vb = nextp(); c.Vw = nextp();
      c.Wb = nextp(); c.Ww = nextp();
    };
    getCellS(ca_); getCellS(cl_); getCellS(cv_);
    o1_b = nextp(); o1_w = nextp(); o2_b = nextp(); o2_w = nextp();
    ra_b = nextp(); ra_w = nextp(); rl_b = nextp(); rl_w = nextp();
    rv_b = nextp(); rv_w = nextp();
  }

  // ----- workspace layout (floats) -----
  float* Wf = (float*)d_ws;
  size_t o_xm = 0;
  size_t o_q = o_xm + (size_t)MROWS * 512;
  size_t o_k = o_q + (size_t)MROWS * HID;
  size_t o_v = o_k + (size_t)MROWS * HID;
  size_t o_o = o_v + (size_t)MROWS * HID;
  size_t o_outl = o_o + (size_t)MROWS * HID;
  size_t o_outa = o_outl + (size_t)MROWS * DM_L;
  size_t o_scan = o_outa + (size_t)MROWS * DM_A;

  float* xm = Wf + o_xm;
  float* Qb = Wf + o_q;
  float* Kb = Wf + o_k;
  float* Vb = Wf + o_v;
  float* Ob = Wf + o_o;  // also FFN-t1 scratch
  float* out_l = Wf + o_outl;
  float* out_a = Wf + o_outa;
  float* out_v = Qb;     // alias: Q/K dead when modality-v FFN output is written

  float* p = Wf + o_scan;
  float *cl0 = p; p += 128 * DH_L; float* cl1 = p; p += 128 * DH_L;
  float *hl0 = p; p += 128 * DH_L; float* hl1 = p; p += 128 * DH_L;
  float *ca0 = p; p += 128 * DH_A; float* ca1 = p; p += 128 * DH_A;
  float *ha0 = p; p += 128 * DH_A; float* ha1 = p; p += 128 * DH_A;
  float *cv0 = p; p += 128 * DH_V; float* cv1 = p; p += 128 * DH_V;
  float *hv0 = p; p += 128 * DH_V; float* hv1 = p; p += 128 * DH_V;
  float *z0 = p; p += 128 * TOT_H; float* z1 = p; p += 128 * TOT_H;
  float* attbuf = p; p += 128 * 832;
  float* redb = p; p += 128 * RSUM;
  float* t1b = p; p += 128 * MAP_H;
  float* ub = p; p += 128 * H_OUT;
  float* WlT = p; p += DM_L * 4 * DH_L;
  float* UlT = p; p += DH_L * 4 * DH_L;
  float* VlT = p; p += TOT_H * 4 * DH_L;
  float* WaT = p; p += DM_A * 4 * DH_A;
  float* UaT = p; p += DH_A * 4 * DH_A;
  float* VaT = p; p += TOT_H * 4 * DH_A;
  float* WvT = p; p += DM_V * 4 * DH_V;
  float* UvT = p; p += DH_V * 4 * DH_V;
  float* VvT = p; p += TOT_H * 4 * DH_V;
  float* attT = p; p += TOT_H * 832;
  float* rlT = p; p += 512 * RL_N;
  float* raT = p; p += 64 * RA_N;
  float* rvT = p; p += 256 * RV_N;
  float* fc1T = p; p += RSUM * MAP_H;
  float* fc2T = p; p += MAP_H * TOT_H;
  float* o1T = p; p += 416 * H_OUT;

  auto tr = [&](const float* W, float* WT, int K, int Nout) {
    int tot = K * Nout;
    transpose_kernel<<<(tot + 255) / 256, 256, 0, stream>>>(W, WT, K, Nout);
  };
  tr(cl_.Ww, WlT, DM_L, 4 * DH_L); tr(cl_.Uw, UlT, DH_L, 4 * DH_L);
  tr(cl_.Vw, VlT, TOT_H, 4 * DH_L);
  tr(ca_.Ww, WaT, DM_A, 4 * DH_A); tr(ca_.Uw, UaT, DH_A, 4 * DH_A);
  tr(ca_.Vw, VaT, TOT_H, 4 * DH_A);
  tr(cv_.Ww, WvT, DM_V, 4 * DH_V); tr(cv_.Uw, UvT, DH_V, 4 * DH_V);
  tr(cv_.Vw, VvT, TOT_H, 4 * DH_V);
  tr(att_w, attT, TOT_H, 832);
  tr(rl_w, rlT, 512, RL_N); tr(ra_w, raT, 64, RA_N); tr(rv_w, rvT, 256, RV_N);
  tr(fc1_w, fc1T, RSUM, MAP_H); tr(fc2_w, fc2T, MAP_H, TOT_H);
  tr(o1_w, o1T, 416, H_OUT);

  auto gemm = [&](const float* Am, const float* Wm, const float* bias,
                  const float* res, float* Y, int K, int Nout, int act) {
    dim3 grid(MROWS / 128, (Nout + 63) / 64);
    gemm_kernel<<<grid, 256, 0, stream>>>(Am, Wm, bias, res, Y, K, Nout, act);
  };
  auto ln = [&](float* Y, const float* g, const float* b, int dm) {
    ln_kernel<<<MROWS / 8, 256, 0, stream>>>(Y, g, b, dm);
  };

  struct Mod { int dm, off; const EncP* e; float* outbuf; };
  Mod mods[3] = { {DM_L, 0, &el, out_l},
                  {DM_A, DM_L, &ea, out_a},
                  {DM_V, DM_L + DM_A, &ev, out_v} };

  const size_t attn_lds =
      (size_t)(256 * KS_STR + DKV * SW_STR + 8 * 16 * KS_STR + 8 * 16 * SW_STR) *
      sizeof(float);

  for (int m = 0; m < 3; ++m) {
    const Mod& md = mods[m];
    int dm = md.dm;
    const EncP& e = *md.e;
    int tot = MROWS * dm;
    slice_kernel<<<(tot + 255) / 256, 256, 0, stream>>>(xptr, xm, dm, md.off);
    gemm(xm, e.wq_w, e.wq_b, nullptr, Qb, dm, HID, 0);
    gemm(xm, e.wk_w, e.wk_b, nullptr, Kb, dm, HID, 0);
    gemm(xm, e.wv_w, e.wv_b, nullptr, Vb, dm, HID, 0);
    attn_kernel<<<N_BATCH * NHEAD, 256, attn_lds, stream>>>(Qb, Kb, Vb, Ob);
    gemm(Ob, e.fc_w, e.fc_b, xm, xm, HID, dm, 0);
    ln(xm, e.ln1_g, e.ln1_b, dm);
    gemm(xm, e.w1_w, e.w1_b, nullptr, Ob, dm, D_INNER, 1);
    gemm(Ob, e.w2_w, e.w2_b, xm, md.outbuf, D_INNER, dm, 0);
    ln(md.outbuf, e.ln2_g, e.ln2_b, dm);
  }

  // ----- persistent recurrent scan -----
  ScanArgs SA;
  SA.xl = out_l; SA.xa = out_a; SA.xv = out_v;
  SA.WlT = WlT; SA.UlT = UlT; SA.VlT = VlT;
  SA.Wl_b = cl_.Wb; SA.Ul_b = cl_.Ub; SA.Vl_b = cl_.Vb;
  SA.WaT = WaT; SA.UaT = UaT; SA.VaT = VaT;
  SA.Wa_b = ca_.Wb; SA.Ua_b = ca_.Ub; SA.Va_b = ca_.Vb;
  SA.WvT = WvT; SA.UvT = UvT; SA.VvT = VvT;
  SA.Wv_b = cv_.Wb; SA.Uv_b = cv_.Ub; SA.Vv_b = cv_.Vb;
  SA.attT = attT; SA.att_b = att_b;
  SA.rlT = rlT; SA.rl_b = rl_b; SA.raT = raT; SA.ra_b = ra_b;
  SA.rvT = rvT; SA.rv_b = rv_b;
  SA.fc1T = fc1T; SA.fc1_b = fc1_b; SA.fc2T = fc2T; SA.fc2_b = fc2_b;
  SA.o1T = o1T; SA.o1_b = o1_b; SA.o2_w = o2_w; SA.o2_b = o2_b;
  SA.cl[0] = cl0; SA.cl[1] = cl1; SA.hl[0] = hl0; SA.hl[1] = hl1;
  SA.ca[0] = ca0; SA.ca[1] = ca1; SA.ha[0] = ha0; SA.ha[1] = ha1;
  SA.cv[0] = cv0; SA.cv[1] = cv1; SA.hv[0] = hv0; SA.hv[1] = hv1;
  SA.z[0] = z0; SA.z[1] = z1;
  SA.attbuf = attbuf; SA.red = redb; SA.t1 = t1b; SA.u = ub;
  SA.out = (float*)d_out;

  scan_kernel<<<1, 1024, 0, stream>>>(SA);
}